// Involution3d_26302379721169
// MI455X (gfx1250) — compile-verified
//
#include <hip/hip_runtime.h>

// Problem constants (match reference)
constexpr int B_    = 2;
constexpr int C_    = 64;
constexpr int G_    = 8;
constexpr int K_    = 27;          // 3*3*3
constexpr int KG_   = 216;         // K*G
constexpr int NSP_  = 32768;       // 32*32*32 spatial positions (od*oh*ow)

typedef __attribute__((ext_vector_type(2))) float v2f;
typedef __attribute__((ext_vector_type(8))) float v8f;

#if __has_builtin(__builtin_amdgcn_wmma_f32_16x16x4_f32)
#define USE_WMMA_F32 1
#endif

// One 16x16 (MxN) f32 tile, K=64, D = A*B with A/B fetched via functors.
// Lane decomposition: l16 = lane%16, lh = lane/16 (wave32).
//   A(m,k): lane = m + 16*(k>=2), vgpr = k%2
//   B(k,n): lane = n + 16*(k>=2), vgpr = k%2
//   D(m,n): vgpr j -> m = j + 8*lh, n = l16
template <typename FA, typename FB>
__device__ __forceinline__ v8f mma_tile_k64(FA fa, FB fb, int l16, int lh) {
  v8f acc = {};
#ifdef USE_WMMA_F32
  #pragma unroll
  for (int k0 = 0; k0 < 64; k0 += 4) {
    const int ka = k0 + 2 * lh;
    v2f a = { fa(l16, ka), fa(l16, ka + 1) };
    v2f b = { fb(ka, l16), fb(ka + 1, l16) };
    acc = __builtin_amdgcn_wmma_f32_16x16x4_f32(
        /*neg_a=*/false, a, /*neg_b=*/false, b,
        /*c_mod=*/(short)0, acc, /*reuse_a=*/false, /*reuse_b=*/false);
  }
#else
  // Scalar fallback honoring the same D layout (compile-safety only).
  #pragma unroll
  for (int j = 0; j < 8; ++j) {
    float s = 0.f;
    for (int k = 0; k < 64; ++k) s += fa(j + 8 * lh, k) * fb(k, l16);
    acc[j] = s;
  }
#endif
  return acc;
}

// Fused: s = relu(BN(w_reduce @ x)) staged in LDS, then ker = w_span @ s -> ws.
// Grid: (NSP/32, B). Block: 256 threads = 8 waves. Each block handles 32 spatial cols.
__global__ __launch_bounds__(256) void involution_kergen(
    const float* __restrict__ x,        // [B, C, NSP]
    const float* __restrict__ w_reduce, // [C, C]
    const float* __restrict__ w_span,   // [KG, C]
    const float* __restrict__ gamma, const float* __restrict__ beta,
    const float* __restrict__ rmean, const float* __restrict__ rvar,
    float* __restrict__ ker)            // [B, KG, NSP]
{
  __shared__ float s_lds[64 * 33];      // 64 channels x 32 cols (stride 33: bank pad)

  const int b    = blockIdx.y;
  const float* xb = x + (size_t)b * C_ * NSP_;
  float* kerb     = ker + (size_t)b * KG_ * NSP_;

  const int tid  = threadIdx.x;
  const int wave = tid >> 5;
  const int lane = tid & 31;
  const int l16  = lane & 15;
  const int lh   = lane >> 4;

  // ---- Phase 1: s tile [64 x 32] ----
  {
    const int mt = wave >> 1, nt = wave & 1;       // 4 M-tiles x 2 N-tiles
    const int m0 = mt * 16;
    const int n0 = blockIdx.x * 32 + nt * 16;
    v8f acc = mma_tile_k64(
        [&](int m, int k) { return w_reduce[(m0 + m) * 64 + k]; },
        [&](int k, int n) { return xb[k * NSP_ + n0 + n]; },
        l16, lh);
    const int nl = nt * 16 + l16;
    #pragma unroll
    for (int j = 0; j < 8; ++j) {
      const int m  = m0 + j + 8 * lh;              // output channel of r
      const float inv = rsqrtf(rvar[m] + 1e-5f);
      float s = (acc[j] - rmean[m]) * inv * gamma[m] + beta[m];
      s_lds[m * 33 + nl] = fmaxf(s, 0.f);
    }
  }
  __syncthreads();

  // ---- Phase 2: ker tile [216(->224) x 32], 28 wave-tiles over 8 waves ----
  for (int t = wave; t < 28; t += 8) {
    const int mt = t >> 1, nt = t & 1;
    const int m0  = mt * 16;
    const int n0l = nt * 16;
    const int n0g = blockIdx.x * 32 + n0l;
    v8f acc = mma_tile_k64(
        [&](int m, int k) { const int row = m0 + m;
                            return row < KG_ ? w_span[row * 64 + k] : 0.f; },
        [&](int k, int n) { return s_lds[k * 33 + n0l + n]; },
        l16, lh);
    #pragma unroll
    for (int j = 0; j < 8; ++j) {
      const int m = m0 + j + 8 * lh;
      if (m < KG_) kerb[(size_t)m * NSP_ + n0g + l16] = acc[j];
    }
  }
}

// out[b,c,sp] = sum_k ker[b, (c/8)*27+k, sp] * x_pad[b, c, nbr(k, sp)]
// with the reference's mixed reshape: flat = k*32768+sp, sp' = flat/27, kid = flat%27.
__global__ __launch_bounds__(256) void involution_gather(
    const float* __restrict__ x,   // [B, C, 32,32,32]
    const float* __restrict__ ker, // [B, KG, NSP]
    float* __restrict__ out)       // [B, C, NSP]
{
  const int idx = blockIdx.x * 256 + threadIdx.x;  // over B*C*NSP
  const int sp  = idx & (NSP_ - 1);
  const int c   = (idx >> 15) & 63;
  const int b   = idx >> 21;

  const float* xb   = x + ((size_t)(b * C_ + c)) * NSP_;
  const float* kerb = ker + ((size_t)(b * KG_ + (c >> 3) * K_)) * NSP_ + sp;

  float acc = 0.f;
  #pragma unroll
  for (int k = 0; k < K_; ++k) {
    if (k + 1 < K_) __builtin_prefetch(kerb + (size_t)(k + 1) * NSP_, 0, 0);
    const float kv = kerb[(size_t)k * NSP_];

    const int flat = (k << 15) + sp;
    const int sp2  = flat / 27;
    const int kid  = flat - sp2 * 27;
    const int dz = sp2 >> 10, dy = (sp2 >> 5) & 31, dx = sp2 & 31;
    const int kd = kid / 9;
    const int r9 = kid - kd * 9;
    const int kh = r9 / 3;
    const int kw = r9 - kh * 3;
    const int zz = dz + kd - 1, yy = dy + kh - 1, xx = dx + kw - 1;

    float xv = 0.f;
    if (((unsigned)zz < 32u) & ((unsigned)yy < 32u) & ((unsigned)xx < 32u))
      xv = xb[(zz << 10) + (yy << 5) + xx];
    acc = fmaf(kv, xv, acc);
  }
  out[idx] = acc;
}

extern "C" void kernel_launch(void* const* d_in, const int* in_sizes, int n_in,
                              void* d_out, int out_size, void* d_ws, size_t ws_size,
                              hipStream_t stream) {
  (void)in_sizes; (void)n_in; (void)out_size; (void)ws_size;
  const float* x        = (const float*)d_in[0];
  const float* w_reduce = (const float*)d_in[1];
  const float* w_span   = (const float*)d_in[2];
  const float* gamma    = (const float*)d_in[3];
  const float* beta     = (const float*)d_in[4];
  const float* rmean    = (const float*)d_in[5];
  const float* rvar     = (const float*)d_in[6];
  float* out            = (float*)d_out;
  float* ker_ws         = (float*)d_ws;   // needs B*KG*NSP*4 = 56.6 MB

  dim3 g1(NSP_ / 32, B_);
  involution_kergen<<<g1, 256, 0, stream>>>(x, w_reduce, w_span, gamma, beta,
                                            rmean, rvar, ker_ws);

  const int total = B_ * C_ * NSP_;      // 4,194,304
  involution_gather<<<total / 256, 256, 0, stream>>>(x, ker_ws, out);
}